// MixHopNet_84507776516697
// MI455X (gfx1250) — compile-verified
//
#include <hip/hip_runtime.h>
#include <math.h>

typedef __attribute__((ext_vector_type(2))) float v2f;
typedef __attribute__((ext_vector_type(8))) float v8f;

// ---------------------------------------------------------------------------
// gcn_norm: deg = 1 (self loop) + count(dst);  dinv = rsqrt(deg)
// ---------------------------------------------------------------------------
__global__ __launch_bounds__(256) void k_deg_init(float* deg, int n) {
  int i = blockIdx.x * 256 + threadIdx.x;
  if (i < n) deg[i] = 1.0f;
}

__global__ __launch_bounds__(256) void k_deg_edges(const int* __restrict__ dst,
                                                   float* deg, int e) {
  int i = blockIdx.x * 256 + threadIdx.x;
  if (i < e) unsafeAtomicAdd(&deg[dst[i]], 1.0f);
}

__global__ __launch_bounds__(256) void k_dinv(float* deg, int n) {
  int i = blockIdx.x * 256 + threadIdx.x;
  if (i < n) deg[i] = rsqrtf(deg[i]);  // deg >= 1 always
}

// ---------------------------------------------------------------------------
// propagate: out = dinv^2 * x   (self-loop term, also initializes out)
// ---------------------------------------------------------------------------
__global__ __launch_bounds__(256) void k_prop_self(const float* __restrict__ x,
                                                   const float* __restrict__ dinv,
                                                   float* __restrict__ out,
                                                   int f4, long total) {
  long i = (long)blockIdx.x * 256 + threadIdx.x;
  if (i >= total) return;
  int node = (int)(i / f4);
  float d = dinv[node];
  float w = d * d;
  float4 v = ((const float4*)x)[i];
  v.x *= w; v.y *= w; v.z *= w; v.w *= w;
  ((float4*)out)[i] = v;
}

// propagate: out[dst] += dinv[src]*dinv[dst] * x[src]   (one wave per edge)
__global__ __launch_bounds__(256) void k_prop_edges(const float* __restrict__ x,
                                                    const int* __restrict__ src,
                                                    const int* __restrict__ dst,
                                                    const float* __restrict__ dinv,
                                                    float* __restrict__ out,
                                                    int e, int f4, int f) {
  int gid = blockIdx.x * 256 + threadIdx.x;
  int edge = gid >> 5;
  int lane = gid & 31;
  if (edge >= e) return;
  int s = src[edge];
  int d = dst[edge];
  float w = dinv[s] * dinv[d];
  const float4* xs = (const float4*)(x + (long)s * f);
  float* od = out + (long)d * f;
  for (int j = lane; j < f4; j += 32) {
    float4 v = xs[j];
    unsafeAtomicAdd(od + j * 4 + 0, v.x * w);
    unsafeAtomicAdd(od + j * 4 + 1, v.y * w);
    unsafeAtomicAdd(od + j * 4 + 2, v.z * w);
    unsafeAtomicAdd(od + j * 4 + 3, v.w * w);
  }
}

// ---------------------------------------------------------------------------
// GEMM1: H[:, colOff:colOff+128] = relu(X[N,128] @ W[128,128] + bias)
// block = 256 thr = 8 waves; block owns 16 rows; wave w owns cols [16w,16w+16)
// K-loop: 32 x v_wmma_f32_16x16x4_f32
// ---------------------------------------------------------------------------
__global__ __launch_bounds__(256) void k_gemm1(const float* __restrict__ X,
                                               const float* __restrict__ W,
                                               const float* __restrict__ bias,
                                               float* __restrict__ H, int colOff) {
  __shared__ float lA[16][132];  // +4 pad -> bank-conflict-free A reads
  int tid = threadIdx.x;
  long rowBase = (long)blockIdx.x * 16;

  // stage 16x128 X tile (contiguous 2048 floats), coalesced float4
  const float4* X4 = (const float4*)(X + rowBase * 128);
  for (int i = tid; i < 512; i += 256) {
    float4 v = X4[i];
    int r = i >> 5;          // 32 float4 per row
    int c = (i & 31) << 2;
    lA[r][c] = v.x; lA[r][c + 1] = v.y; lA[r][c + 2] = v.z; lA[r][c + 3] = v.w;
  }
  __syncthreads();

  int wv = tid >> 5;
  int lane = tid & 31;
  int n = lane & 15;         // A: row; B/C/D: column
  int hi = lane >> 4;        // A: K subgroup; C/D: row subgroup (M += 8*hi)
  int colBase = wv << 4;

  v8f acc = {0.f, 0.f, 0.f, 0.f, 0.f, 0.f, 0.f, 0.f};
  for (int k0 = 0; k0 < 128; k0 += 4) {
    int k = k0 + 2 * hi;
    v2f a;
    a.x = lA[n][k];
    a.y = lA[n][k + 1];
    v2f b;
    b.x = W[(long)k * 128 + colBase + n];
    b.y = W[(long)(k + 1) * 128 + colBase + n];
    acc = __builtin_amdgcn_wmma_f32_16x16x4_f32(false, a, false, b,
                                                (short)0, acc, false, false);
  }

  float bv = bias[colBase + n];
  float* Hp = H + (rowBase + hi * 8) * 384 + colOff + colBase + n;
#pragma unroll
  for (int r = 0; r < 8; ++r) {
    float v = acc[r] + bv;
    Hp[(long)r * 384] = fmaxf(v, 0.0f);
  }
}

// ---------------------------------------------------------------------------
// GEMM2: out[:,0:40]=h@W0+b0 ; out[:,40:80]=h1@W1+b1  (cols padded 40->48)
// block = 192 thr = 6 waves: part = wv/3 (h vs h1), colTile = wv%3
// ---------------------------------------------------------------------------
__global__ __launch_bounds__(192) void k_gemm2(const float* __restrict__ Ha,
                                               const float* __restrict__ Hb,
                                               const float* __restrict__ W0,
                                               const float* __restrict__ b0,
                                               const float* __restrict__ W1,
                                               const float* __restrict__ b1,
                                               float* __restrict__ OUT) {
  __shared__ float lA[2][16][388];  // +4 pad
  int tid = threadIdx.x;
  long rowBase = (long)blockIdx.x * 16;

  for (int p = 0; p < 2; ++p) {
    const float* S = p ? Hb : Ha;
    const float4* S4 = (const float4*)(S + rowBase * 384);
    for (int i = tid; i < 1536; i += 192) {
      float4 v = S4[i];
      int r = i / 96;        // 96 float4 per row
      int c = (i % 96) << 2;
      lA[p][r][c] = v.x; lA[p][r][c + 1] = v.y;
      lA[p][r][c + 2] = v.z; lA[p][r][c + 3] = v.w;
    }
  }
  __syncthreads();

  int wv = tid >> 5;             // 0..5
  int lane = tid & 31;
  int part = wv / 3;             // 0: h@W0, 1: h1@W1
  int colTile = wv % 3;          // cols [16t, 16t+16) of the 40(+pad) outputs
  const float* W = part ? W1 : W0;
  const float* bb = part ? b1 : b0;
  int n = lane & 15;
  int hi = lane >> 4;
  int col = colTile * 16 + n;    // 0..47
  bool valid = col < 40;

  v8f acc = {0.f, 0.f, 0.f, 0.f, 0.f, 0.f, 0.f, 0.f};
  for (int k0 = 0; k0 < 384; k0 += 4) {
    int k = k0 + 2 * hi;
    v2f a;
    a.x = lA[part][n][k];
    a.y = lA[part][n][k + 1];
    v2f b;
    b.x = valid ? W[(long)k * 40 + col] : 0.0f;       // reconverges before wmma
    b.y = valid ? W[(long)(k + 1) * 40 + col] : 0.0f;
    acc = __builtin_amdgcn_wmma_f32_16x16x4_f32(false, a, false, b,
                                                (short)0, acc, false, false);
  }

  if (valid) {
    float bv = bb[col];
    float* O = OUT + (rowBase + hi * 8) * 80 + part * 40 + col;
#pragma unroll
    for (int r = 0; r < 8; ++r) O[(long)r * 80] = acc[r] + bv;
  }
}

// ---------------------------------------------------------------------------
// log_softmax over rows of [N,80], in place. One wave (32 lanes) per row.
// ---------------------------------------------------------------------------
__global__ __launch_bounds__(256) void k_logsoftmax(float* __restrict__ OUT, int n) {
  int gid = blockIdx.x * 256 + threadIdx.x;
  int row = gid >> 5;
  int lane = gid & 31;
  if (row >= n) return;
  float* o = OUT + (long)row * 80;

  float v0 = o[lane];
  float v1 = o[lane + 32];
  float v2 = (lane < 16) ? o[lane + 64] : -1e30f;

  float m = fmaxf(fmaxf(v0, v1), v2);
#pragma unroll
  for (int off = 16; off > 0; off >>= 1) m = fmaxf(m, __shfl_xor(m, off));

  float s = expf(v0 - m) + expf(v1 - m) + ((lane < 16) ? expf(v2 - m) : 0.0f);
#pragma unroll
  for (int off = 16; off > 0; off >>= 1) s += __shfl_xor(s, off);

  float ls = m + logf(s);
  o[lane] = v0 - ls;
  o[lane + 32] = v1 - ls;
  if (lane < 16) o[lane + 64] = v2 - ls;
}

// ---------------------------------------------------------------------------
extern "C" void kernel_launch(void* const* d_in, const int* in_sizes, int n_in,
                              void* d_out, int out_size, void* d_ws, size_t ws_size,
                              hipStream_t stream) {
  const float* x    = (const float*)d_in[0];
  const int*   ei   = (const int*)d_in[1];
  const float* w1_0 = (const float*)d_in[2];
  const float* b1_0 = (const float*)d_in[3];
  const float* w1_1 = (const float*)d_in[4];
  const float* b1_1 = (const float*)d_in[5];
  const float* w1_2 = (const float*)d_in[6];
  const float* b1_2 = (const float*)d_in[7];
  const float* w2_0 = (const float*)d_in[8];
  const float* b2_0 = (const float*)d_in[9];
  const float* w2_1 = (const float*)d_in[10];
  const float* b2_1 = (const float*)d_in[11];
  float* out = (float*)d_out;

  const int N = in_sizes[0] / 128;   // 50000 (divisible by 16)
  const int E = in_sizes[1] / 2;     // 800000
  const int* src = ei;               // edge_index[0]
  const int* dst = ei + E;           // edge_index[1]

  float* wsf  = (float*)d_ws;
  float* dinv = wsf;
  float* x1   = dinv + N;
  float* x2   = x1 + (long)N * 128;
  float* h    = x2 + (long)N * 128;
  float* h1   = h + (long)N * 384;

  int nb  = (N + 255) / 256;
  int eb  = (E + 255) / 256;
  int ewb = (E * 32 + 255) / 256;    // one wave per edge

  // gcn_norm
  k_deg_init<<<nb, 256, 0, stream>>>(dinv, N);
  k_deg_edges<<<eb, 256, 0, stream>>>(dst, dinv, E);
  k_dinv<<<nb, 256, 0, stream>>>(dinv, N);

  long t128 = (long)N * 32;   // float4 count, F=128
  long t384 = (long)N * 96;   // float4 count, F=384
  int pb128 = (int)((t128 + 255) / 256);
  int pb384 = (int)((t384 + 255) / 256);

  // x1 = A x ; x2 = A x1
  k_prop_self<<<pb128, 256, 0, stream>>>(x, dinv, x1, 32, t128);
  k_prop_edges<<<ewb, 256, 0, stream>>>(x, src, dst, dinv, x1, E, 32, 128);
  k_prop_self<<<pb128, 256, 0, stream>>>(x1, dinv, x2, 32, t128);
  k_prop_edges<<<ewb, 256, 0, stream>>>(x1, src, dst, dinv, x2, E, 32, 128);

  // h = relu([x@W0 | x1@W1 | x2@W2] + b)
  int rt = N / 16;  // 3125 row tiles
  k_gemm1<<<rt, 256, 0, stream>>>(x,  w1_0, b1_0, h, 0);
  k_gemm1<<<rt, 256, 0, stream>>>(x1, w1_1, b1_1, h, 128);
  k_gemm1<<<rt, 256, 0, stream>>>(x2, w1_2, b1_2, h, 256);

  // h1 = A h
  k_prop_self<<<pb384, 256, 0, stream>>>(h, dinv, h1, 96, t384);
  k_prop_edges<<<ewb, 256, 0, stream>>>(h, src, dst, dinv, h1, E, 96, 384);

  // out = [h@W2_0+b | h1@W2_1+b] ; then log_softmax rows
  k_gemm2<<<rt, 192, 0, stream>>>(h, h1, w2_0, b2_0, w2_1, b2_1, out);
  k_logsoftmax<<<(N + 7) / 8, 256, 0, stream>>>(out, N);
}